// ExpertChoiceRouter_2018634629602
// MI455X (gfx1250) — compile-verified
//
#include <hip/hip_runtime.h>

// Problem constants (match reference)
#define B_    4
#define S_    4096
#define D_    2048
#define E_    64
#define KTOP_ 128
#define T_    (B_ * S_)        // 16384 tokens
#define N_    (T_ * E_)        // 1048576 elements per output tensor

typedef float v8f __attribute__((ext_vector_type(8)));
typedef float v2f __attribute__((ext_vector_type(2)));

#define KC      64             // K-chunk staged in LDS (double-buffered)
#define NCHUNK  (D_ / KC)      // 32
#define LSTRIDE 68             // padded row stride (floats): 16B-aligned rows, conflict-free reads

// --------------------------- async LDS copy path ---------------------------
#if defined(__gfx1250__) && __has_builtin(__builtin_amdgcn_global_load_async_to_lds_b128)
#define HAVE_ASYNC_LDS 1
#else
#define HAVE_ASYNC_LDS 0
#endif

#if HAVE_ASYNC_LDS
typedef int i4v __attribute__((vector_size(16)));
typedef __attribute__((address_space(1))) i4v* gi4p;   // global int32x4*
typedef __attribute__((address_space(3))) i4v* li4p;   // LDS    int32x4*
typedef __attribute__((address_space(1))) void* gvp;
typedef __attribute__((address_space(3))) void* lvp;
#endif

// Copy 16 bytes global -> LDS. Async path uses GLOBAL_LOAD_ASYNC_TO_LDS_B128
// (no VGPR data round-trip, tracked by ASYNCcnt).
__device__ __forceinline__ void copy16_g2l(const float* g, float* l) {
#if HAVE_ASYNC_LDS
  __builtin_amdgcn_global_load_async_to_lds_b128(
      (gi4p)(gvp)(void*)g,
      (li4p)(lvp)(void*)l,
      /*imm offset=*/0, /*cpol=*/0);
#else
  *(float4*)l = *(const float4*)g;
#endif
}

__device__ __forceinline__ void wait_async_copies() {
#if HAVE_ASYNC_LDS
#if __has_builtin(__builtin_amdgcn_s_wait_asynccnt)
  __builtin_amdgcn_s_wait_asynccnt(0);
#else
  asm volatile("s_wait_asynccnt 0" ::: "memory");
#endif
#endif
}

// ---------------------------------------------------------------------------
// Kernel 1: logits = X (T x D) * W^T (D x E) via V_WMMA_F32_16X16X4_F32.
// Each wave: 16 tokens x 64 experts (4 accumulator tiles). 4 waves / block.
// Double-buffered LDS staging with async global->LDS copies overlapping WMMA.
// ---------------------------------------------------------------------------
__global__ __launch_bounds__(128) void ecr_gemm_logits(
    const float* __restrict__ x, const float* __restrict__ W,
    float* __restrict__ logits, float* __restrict__ logitsT)
{
  __shared__ float Xs[2][64 * LSTRIDE];   // 2 x 17 KB
  __shared__ float Ws[2][64 * LSTRIDE];   // 2 x 17 KB  (total ~68 KB)

  const int tid  = threadIdx.x;
  const int wave = tid >> 5;
  const int lane = tid & 31;
  const int half = lane >> 4;             // 0: K0/K1, 1: K2/K3 (ISA fp32 A/B striping)
  const int m    = lane & 15;
  const int tok0 = blockIdx.x * 64;

  // Issue async copies for one 64x64 K-chunk of X and W into LDS buffer `buf`.
  auto stage = [&](int kc, int buf) {
    #pragma unroll
    for (int i = 0; i < 8; ++i) {
      int f4  = tid + i * 128;            // 0..1023 (float4 index in 64x64 tile)
      int row = f4 >> 4;                  // 16 float4 per row
      int c4  = f4 & 15;
      const float* gx = x + ((size_t)(tok0 + row) * D_ + kc * KC + c4 * 4);
      const float* gw = W + ((size_t)row * D_ + kc * KC + c4 * 4);
      copy16_g2l(gx, &Xs[buf][row * LSTRIDE + c4 * 4]);
      copy16_g2l(gw, &Ws[buf][row * LSTRIDE + c4 * 4]);
    }
  };

  v8f acc[4];
  v8f vzero = {0.f, 0.f, 0.f, 0.f, 0.f, 0.f, 0.f, 0.f};
  acc[0] = vzero; acc[1] = vzero; acc[2] = vzero; acc[3] = vzero;

  const int arow = (wave * 16 + m) * LSTRIDE + 2 * half;
  const int boff = m * LSTRIDE + 2 * half;

  // Prologue: fill buffer 0.
  stage(0, 0);
  wait_async_copies();
  __syncthreads();

  for (int kc = 0; kc < NCHUNK; ++kc) {
    const int cur = kc & 1;
    // Kick off next chunk's async copies into the other buffer, then compute.
    if (kc + 1 < NCHUNK) stage(kc + 1, cur ^ 1);

    const float* xsb = &Xs[cur][0];
    const float* wsb = &Ws[cur][0];
    #pragma unroll 4
    for (int kk = 0; kk < KC; kk += 4) {
      // A tile (16x4 f32): lane m holds row m; v0/v1 = K0/K1 (lanes 0-15), K2/K3 (lanes 16-31)
      v2f a = *(const v2f*)&xsb[arow + kk];
      #pragma unroll
      for (int nt = 0; nt < 4; ++nt) {
        // B tile (4x16 f32): lane n holds column n = expert nt*16+m
        v2f bb = *(const v2f*)&wsb[nt * 16 * LSTRIDE + boff + kk];
        acc[nt] = __builtin_amdgcn_wmma_f32_16x16x4_f32(
            /*neg_a=*/false, a, /*neg_b=*/false, bb,
            /*c_mod=*/(short)0, acc[nt], /*reuse_a=*/false, /*reuse_b=*/false);
      }
    }

    // Drain this wave's async copies, then workgroup barrier -> next buffer ready.
    wait_async_copies();
    __syncthreads();
  }

  // C/D layout: VGPR j holds M=j (lanes 0-15) and M=j+8 (lanes 16-31), N = lane&15.
  const int tokBase = tok0 + wave * 16 + half * 8;
  #pragma unroll
  for (int nt = 0; nt < 4; ++nt) {
    const int e = nt * 16 + m;
    #pragma unroll
    for (int j = 0; j < 8; ++j) {
      int   tok = tokBase + j;
      float v   = acc[nt][j];
      logits[(size_t)tok * E_ + e] = v;
      int b = tok >> 12;               // S_ == 4096
      int s = tok & (S_ - 1);
      logitsT[((size_t)b * E_ + e) * S_ + s] = v;
    }
  }
}

// Order-preserving float -> uint key (monotone, handles negatives).
__device__ __forceinline__ unsigned ordkey(float f) {
  unsigned u = __float_as_uint(f);
  return (u & 0x80000000u) ? ~u : (u | 0x80000000u);
}

// ---------------------------------------------------------------------------
// Kernel 2: per (b,e): softmax stats over S, plus exact top-K threshold
// (K-th largest logit key + index-ordered tie cutoff).
// One 256-thread block per (b,e); each thread owns 16 consecutive s values.
// ---------------------------------------------------------------------------
__global__ __launch_bounds__(256) void ecr_softmax_topk_stats(
    const float* __restrict__ logitsT,
    float* __restrict__ maxArr, float* __restrict__ invSumArr,
    unsigned* __restrict__ kthArr, int* __restrict__ scutArr)
{
  const int be  = blockIdx.x;          // 0..B*E-1
  const int tid = threadIdx.x;         // 0..255
  const float* row = logitsT + (size_t)be * S_;

  float    vals[16];
  unsigned keys[16];
  const float4* r4 = (const float4*)row;
  #pragma unroll
  for (int i = 0; i < 4; ++i) {
    float4 q = r4[tid * 4 + i];
    vals[i * 4 + 0] = q.x; vals[i * 4 + 1] = q.y;
    vals[i * 4 + 2] = q.z; vals[i * 4 + 3] = q.w;
  }
  #pragma unroll
  for (int i = 0; i < 16; ++i) keys[i] = ordkey(vals[i]);

  __shared__ float red[256];
  __shared__ int   cnt_sh;
  __shared__ int   pre_sh[257];
  __shared__ int   scut_sh;

  // max over S
  float lmax = vals[0];
  #pragma unroll
  for (int i = 1; i < 16; ++i) lmax = fmaxf(lmax, vals[i]);
  red[tid] = lmax; __syncthreads();
  for (int off = 128; off > 0; off >>= 1) {
    if (tid < off) red[tid] = fmaxf(red[tid], red[tid + off]);
    __syncthreads();
  }
  float rmax = red[0]; __syncthreads();

  // sum exp(l - max)
  float lsum = 0.f;
  #pragma unroll
  for (int i = 0; i < 16; ++i) lsum += expf(vals[i] - rmax);
  red[tid] = lsum; __syncthreads();
  for (int off = 128; off > 0; off >>= 1) {
    if (tid < off) red[tid] += red[tid + off];
    __syncthreads();
  }
  float rsum = red[0]; __syncthreads();

  // count of keys >= t (uniform across block; barriers are uniform)
  auto count_ge = [&](unsigned t) -> int {
    if (tid == 0) cnt_sh = 0;
    __syncthreads();
    int c = 0;
    #pragma unroll
    for (int i = 0; i < 16; ++i) c += (keys[i] >= t) ? 1 : 0;
    if (c) atomicAdd(&cnt_sh, c);
    __syncthreads();
    int r = cnt_sh;
    __syncthreads();
    return r;
  };

  // bisection for the K-th largest key: max t with count_ge(t) >= K
  unsigned lo = 0u, hi = 0xFFFFFFFFu;
  while (lo < hi) {
    unsigned span = hi - lo;
    unsigned mid  = lo + (span >> 1) + (span & 1u);   // ceil midpoint, >= lo+1
    int c = count_ge(mid);
    if (c >= KTOP_) lo = mid; else hi = mid - 1u;
  }
  const unsigned kth = lo;
  const int G    = (kth == 0xFFFFFFFFu) ? 0 : count_ge(kth + 1u);  // strictly greater
  const int need = KTOP_ - G;                                      // equals to select (>=1)

  // index-ordered tie cutoff: s of the need-th (0-based) element equal to kth
  int ec = 0;
  #pragma unroll
  for (int i = 0; i < 16; ++i) ec += (keys[i] == kth) ? 1 : 0;
  pre_sh[tid] = ec;
  if (tid == 0) scut_sh = S_;          // default: all equals selected
  __syncthreads();
  if (tid == 0) {
    int a = 0;
    for (int t2 = 0; t2 < 256; ++t2) { int c = pre_sh[t2]; pre_sh[t2] = a; a += c; }
    pre_sh[256] = a;
  }
  __syncthreads();
  int base = pre_sh[tid];
  if (need >= base && need < base + ec) {
    int tgt = need - base, seen = 0;
    for (int i = 0; i < 16; ++i) {
      if (keys[i] == kth) {
        if (seen == tgt) { scut_sh = tid * 16 + i; break; }
        ++seen;
      }
    }
  }
  __syncthreads();

  if (tid == 0) {
    maxArr[be]    = rmax;
    invSumArr[be] = 1.0f / rsum;
    kthArr[be]    = kth;
    scutArr[be]   = scut_sh;
  }
}

// ---------------------------------------------------------------------------
// Kernel 3: probs = exp(l - max) * inv_sum;  mask = selected(top-K) ? 1 : 0
// ---------------------------------------------------------------------------
__global__ __launch_bounds__(256) void ecr_finalize(
    const float* __restrict__ logits,
    const float* __restrict__ maxArr, const float* __restrict__ invSumArr,
    const unsigned* __restrict__ kthArr, const int* __restrict__ scutArr,
    float* __restrict__ mask, float* __restrict__ probs)
{
  const int idx = blockIdx.x * 256 + threadIdx.x;   // < N_
  float l  = logits[idx];
  int e    = idx & (E_ - 1);
  int tok  = idx >> 6;                               // E_ == 64
  int b    = tok >> 12;                              // S_ == 4096
  int s    = tok & (S_ - 1);
  int be   = b * E_ + e;

  float p = expf(l - maxArr[be]) * invSumArr[be];
  probs[idx] = p;

  unsigned key = ordkey(l);
  unsigned kv  = kthArr[be];
  float sel = (key > kv || (key == kv && s < scutArr[be])) ? 1.0f : 0.0f;
  mask[idx] = sel;
}

// ---------------------------------------------------------------------------
extern "C" void kernel_launch(void* const* d_in, const int* in_sizes, int n_in,
                              void* d_out, int out_size, void* d_ws, size_t ws_size,
                              hipStream_t stream) {
  (void)in_sizes; (void)n_in; (void)out_size; (void)ws_size;

  const float* x = (const float*)d_in[0];   // (B,S,D) fp32
  const float* W = (const float*)d_in[1];   // (E,D)   fp32

  float* out    = (float*)d_out;
  float* mask   = out;                      // (B,S,E)
  float* probs  = out + (size_t)N_;         // (B,S,E)
  float* logits = out + (size_t)2 * N_;     // (B,S,E)

  float* wsf      = (float*)d_ws;
  float* logitsT  = wsf;                    // (B,E,S) transposed logits, 4 MB
  float* maxArr   = wsf + (size_t)N_;
  float* invSum   = maxArr + (B_ * E_);
  unsigned* kth   = (unsigned*)(invSum + (B_ * E_));
  int* scut       = (int*)(kth + (B_ * E_));

  ecr_gemm_logits<<<T_ / 64, 128, 0, stream>>>(x, W, logits, logitsT);
  ecr_softmax_topk_stats<<<B_ * E_, 256, 0, stream>>>(logitsT, maxArr, invSum, kth, scut);
  ecr_finalize<<<N_ / 256, 256, 0, stream>>>(logits, maxArr, invSum, kth, scut, mask, probs);
}